// RWKVCNN_Quan_61589831025144
// MI455X (gfx1250) — compile-verified
//
#include <hip/hip_runtime.h>
#include <hip/hip_bf16.h>
#include <stdint.h>

// ---------------- problem constants (match reference) ----------------
static constexpr int Bn = 8, Tn = 1024, Cn = 2048;
static constexpr int PP_INIT = -(1 << 15);
static constexpr long long AA_LO = -(1ll << 23);
static constexpr long long AA_HI =  (1ll << 23) - 1;
static constexpr long long BB_HI =  (1ll << 24) - 1;

// ---------------- tuning ----------------
#define BLOCK   64     // threads per block (2 waves)
#define STAGES  16     // async pipeline depth (power of 2); 4 loads/stage/wave
#define CPT     2      // channels per thread (ILP for the serial chain)
// steady-state: wait ASYNCcnt <= 56, then issue 4 -> peak 60 < 63 (6-bit counter)
// 56 <= 4*(STAGES-1)=60, i.e. strictly stronger than needed for stage-t completion.

#ifndef __has_builtin
#define __has_builtin(x) 0
#endif

#if __has_builtin(__builtin_amdgcn_global_load_async_to_lds_b32)
#define HAVE_ASYNC_LDS 1
#endif

#if __has_builtin(__builtin_amdgcn_s_wait_asynccnt)
#define WAIT_ASYNC(n) __builtin_amdgcn_s_wait_asynccnt(n)
#else
#define WAIT_ASYNC(n) asm volatile("s_wait_asynccnt " #n ::: "memory")
#endif

// ---------------- exact fixed-point helpers ----------------

// LUT index: clip(div_rne(delta - MIN_DELTA_I, 24), 0, 255), MIN_DELTA_I = -6144.
// delta <= 0 always. For x = delta + 6144 <= 0 the RNE quotient is <= 0 -> clips to 0.
__device__ __forceinline__ int lutIdx(int delta) {
    int x = delta + 6144;
    if (x <= 0) return 0;
    unsigned ux = (unsigned)x;
    unsigned q = ux / 24u;              // constant division -> mul-high
    unsigned r = ux - q * 24u;
    unsigned r2 = r << 1;
    q += (unsigned)((r2 > 24u) | ((r2 == 24u) & (q & 1u)));
    return (int)(q > 255u ? 255u : q);
}

// round-to-nearest-even arithmetic >>16, sign-magnitude (|v| < 2^40)
__device__ __forceinline__ long long rshiftRne16(long long v) {
    unsigned long long a = (unsigned long long)(v < 0 ? -v : v);
    unsigned long long q = a >> 16;
    unsigned r = (unsigned)a & 0xFFFFu;
    q += (unsigned long long)((r > 0x8000u) | ((r == 0x8000u) & ((unsigned)q & 1u)));
    return v < 0 ? -(long long)q : (long long)q;
}

// round-to-nearest-even division, sign-magnitude. |a| <= 2^23, 1 <= d < 2^24,
// so the whole thing fits in fast unsigned 32-bit arithmetic.
__device__ __forceinline__ int divRne(long long a, unsigned d) {
    unsigned ua = (unsigned)(a < 0 ? -(a) : a);
    unsigned q = ua / d;
    unsigned r = ua - q * d;
    unsigned r2 = r << 1;               // r < d < 2^24 -> no overflow
    q += (unsigned)((r2 > d) | ((r2 == d) & (q & 1u)));
    int s = (int)q;
    return a < 0 ? -s : s;
}

struct St { int pp; long long aa; long long bb; };

// One full WKV timestep for one channel. Returns y_t.
__device__ __forceinline__ int stepOne(St& s, int kk, int vv, int u, int w,
                                       const int* __restrict__ lut /* LDS */) {
    // output half-step
    int ww = kk + u;
    int p  = s.pp > ww ? s.pp : ww;
    long long e1 = lut[lutIdx(s.pp - p)];
    long long e2 = lut[lutIdx(ww   - p)];
    long long aat = rshiftRne16(s.aa * e1) + (long long)vv * e2;
    aat = aat < AA_LO ? AA_LO : (aat > AA_HI ? AA_HI : aat);
    long long bbt = rshiftRne16(s.bb * e1) + e2;        // >= 0 by construction
    bbt = bbt > BB_HI ? BB_HI : bbt;
    unsigned d = (unsigned)(bbt < 1 ? 1 : bbt);
    int yi = divRne(aat, d);
    // state half-step
    int ww2 = s.pp + w;
    int p2  = ww2 > kk ? ww2 : kk;
    long long e1n = lut[lutIdx(ww2 - p2)];
    long long e2n = lut[lutIdx(kk  - p2)];
    long long aan = rshiftRne16(aat * e1n) + (long long)vv * e2n;
    aan = aan < AA_LO ? AA_LO : (aan > AA_HI ? AA_HI : aan);
    long long bbn = rshiftRne16(bbt * e1n) + e2n;
    bbn = bbn > BB_HI ? BB_HI : bbn;
    s.pp = p2; s.aa = aan; s.bb = bbn;
    return yi;
}

#if defined(HAVE_ASYNC_LDS)
__device__ __forceinline__ void async_ld(const int* g, int* l) {
    __builtin_amdgcn_global_load_async_to_lds_b32(
        (__attribute__((address_space(1))) int*)(g),
        (__attribute__((address_space(3))) int*)(l),
        0, 0);
}
#endif

__global__ __launch_bounds__(BLOCK)
void wkv_int_kernel(const int* __restrict__ w_i, const int* __restrict__ u_i,
                    const int* __restrict__ k_i, const int* __restrict__ v_i,
                    const int* __restrict__ lut_g, int* __restrict__ y) {
    __shared__ int lut[256];
#if defined(HAVE_ASYNC_LDS)
    __shared__ int tiles[STAGES * 4 * BLOCK];   // per stage: k0|v0|k1|v1, BLOCK each
#endif
    const int tid = threadIdx.x;
    for (int i = tid; i < 256; i += BLOCK) lut[i] = lut_g[i];
    __syncthreads();

    const int groups = Cn / (CPT * BLOCK);          // 16
    const int b  = (int)blockIdx.x / groups;
    const int cg = (int)blockIdx.x % groups;
    const int c0 = cg * (CPT * BLOCK) + tid;
    const int c1 = c0 + BLOCK;

    const int u0 = u_i[c0], u1 = u_i[c1];
    const int w0 = w_i[c0], w1 = w_i[c1];

    const size_t base = (size_t)b * Tn * Cn;
    const int* kp0 = k_i + base + c0;
    const int* kp1 = k_i + base + c1;
    const int* vp0 = v_i + base + c0;
    const int* vp1 = v_i + base + c1;
    int* yp0 = y + base + c0;
    int* yp1 = y + base + c1;

    St s0, s1;
    s0.pp = PP_INIT; s0.aa = 0; s0.bb = 0;
    s1.pp = PP_INIT; s1.aa = 0; s1.bb = 0;

#if defined(HAVE_ASYNC_LDS)
    // ---- prologue: fill pipeline (4 async loads per wave per timestep) ----
    for (int s = 0; s < STAGES; ++s) {
        int off = s * 4 * BLOCK + tid;
        size_t go = (size_t)s * Cn;
        async_ld(kp0 + go, &tiles[off]);
        async_ld(vp0 + go, &tiles[off + BLOCK]);
        async_ld(kp1 + go, &tiles[off + 2 * BLOCK]);
        async_ld(vp1 + go, &tiles[off + 3 * BLOCK]);
    }
    // ---- main loop: stage t is complete once <= 4*(STAGES-1)=60 remain in
    // flight; we wait to <= 56 (strictly stronger) to keep peak outstanding
    // at 60 < 63 (6-bit ASYNCcnt) after issuing this iteration's 4 refills ----
    for (int t = 0; t < Tn - STAGES; ++t) {
        WAIT_ASYNC(56);
        int off = (t & (STAGES - 1)) * 4 * BLOCK + tid;
        int kk0 = tiles[off];
        int vv0 = tiles[off + BLOCK];
        int kk1 = tiles[off + 2 * BLOCK];
        int vv1 = tiles[off + 3 * BLOCK];
        int y0 = stepOne(s0, kk0, vv0, u0, w0, lut);
        int y1 = stepOne(s1, kk1, vv1, u1, w1, lut);
        size_t to = (size_t)t * Cn;
        yp0[to] = y0;
        yp1[to] = y1;
        asm volatile("" ::: "memory");  // keep the slot-refill below the ds_loads above
        size_t go = (size_t)(t + STAGES) * Cn;
        async_ld(kp0 + go, &tiles[off]);
        async_ld(vp0 + go, &tiles[off + BLOCK]);
        async_ld(kp1 + go, &tiles[off + 2 * BLOCK]);
        async_ld(vp1 + go, &tiles[off + 3 * BLOCK]);
    }
    // ---- epilogue: drain ----
    WAIT_ASYNC(0);
    for (int t = Tn - STAGES; t < Tn; ++t) {
        int off = (t & (STAGES - 1)) * 4 * BLOCK + tid;
        int kk0 = tiles[off];
        int vv0 = tiles[off + BLOCK];
        int kk1 = tiles[off + 2 * BLOCK];
        int vv1 = tiles[off + 3 * BLOCK];
        int y0 = stepOne(s0, kk0, vv0, u0, w0, lut);
        int y1 = stepOne(s1, kk1, vv1, u1, w1, lut);
        size_t to = (size_t)t * Cn;
        yp0[to] = y0;
        yp1[to] = y1;
    }
#else
    // ---- fallback: register software pipelining with plain global loads ----
    int kk0 = kp0[0], vv0 = vp0[0], kk1 = kp1[0], vv1 = vp1[0];
    for (int t = 0; t < Tn; ++t) {
        int nk0 = 0, nv0 = 0, nk1 = 0, nv1 = 0;
        if (t + 1 < Tn) {
            size_t go = (size_t)(t + 1) * Cn;
            nk0 = kp0[go]; nv0 = vp0[go]; nk1 = kp1[go]; nv1 = vp1[go];
        }
        int y0 = stepOne(s0, kk0, vv0, u0, w0, lut);
        int y1 = stepOne(s1, kk1, vv1, u1, w1, lut);
        size_t to = (size_t)t * Cn;
        yp0[to] = y0;
        yp1[to] = y1;
        kk0 = nk0; vv0 = nv0; kk1 = nk1; vv1 = nv1;
    }
#endif
}

extern "C" void kernel_launch(void* const* d_in, const int* in_sizes, int n_in,
                              void* d_out, int out_size, void* d_ws, size_t ws_size,
                              hipStream_t stream) {
    const int* w_i = (const int*)d_in[0];
    const int* u_i = (const int*)d_in[1];
    const int* k_i = (const int*)d_in[2];
    const int* v_i = (const int*)d_in[3];
    const int* lut = (const int*)d_in[4];
    int* out = (int*)d_out;
    (void)in_sizes; (void)n_in; (void)out_size; (void)d_ws; (void)ws_size;

    dim3 grid(Bn * (Cn / (CPT * BLOCK)));   // 8 * 16 = 128 blocks
    wkv_int_kernel<<<grid, BLOCK, 0, stream>>>(w_i, u_i, k_i, v_i, lut, out);
}